// Model_40870908788924
// MI455X (gfx1250) — compile-verified
//
#include <hip/hip_runtime.h>

// CDNA5 (gfx1250, wave32) implementation of a 2-layer hetero GraphSAGE.
// GEMMs use V_WMMA_F32_16X16X4_F32 (fp32 A/B, fp32 acc): the workload is
// scatter/gather-bandwidth bound (~3 GB @ 23.3 TB/s), so precision-safe fp32
// WMMA is the right choice — matrix throughput is not the bottleneck.
// Scatter-adds use __hip_atomic_fetch_add (relaxed, agent scope) so the 64M
// float atomics lower to native global_atomic_add_f32 (non-returning,
// STOREcnt-tracked) instead of a CAS retry loop.

typedef __attribute__((ext_vector_type(2))) float v2f;
typedef __attribute__((ext_vector_type(8))) float v8f;

constexpr int HDIM = 128;

__device__ __forceinline__ void atomic_add_f32(float* p, float v) {
  (void)__hip_atomic_fetch_add(p, v, __ATOMIC_RELAXED,
                               __HIP_MEMORY_SCOPE_AGENT);
}

// ---------------------------------------------------------------- zero fill
__global__ void zero_kernel(float* __restrict__ p, int n) {
  int i = blockIdx.x * blockDim.x + threadIdx.x;
  int stride = gridDim.x * blockDim.x;
  for (; i < n; i += stride) p[i] = 0.0f;
}

// ------------------------------------------------------------ degree counts
__global__ void degree_kernel(const int* __restrict__ dst, int e,
                              float* __restrict__ deg) {
  int i = blockIdx.x * blockDim.x + threadIdx.x;
  if (i < e) atomic_add_f32(&deg[dst[i]], 1.0f);
}

// ------------------------------------------- scatter-add: s[dst] += x[src]
// One wave32 per edge; each lane moves 4 contiguous floats (128 = 32*4).
// x (2-4 MB) is L2-resident, so the random gather mostly hits L2.
__global__ void scatter_kernel(const float* __restrict__ x,
                               const int* __restrict__ src,
                               const int* __restrict__ dst, int e,
                               float* __restrict__ s) {
  int wid  = (blockIdx.x * blockDim.x + threadIdx.x) >> 5;
  int lane = threadIdx.x & 31;
  if (wid >= e) return;
  int sr = src[wid];
  int dr = dst[wid];
  const float4 v = *(const float4*)(x + (size_t)sr * HDIM + lane * 4);
  float* o = s + (size_t)dr * HDIM + lane * 4;
  atomic_add_f32(o + 0, v.x);
  atomic_add_f32(o + 1, v.y);
  atomic_add_f32(o + 2, v.z);
  atomic_add_f32(o + 3, v.w);
}

// --------------------------------------------------------------------------
// WMMA fragment conventions (ISA §7.12.2, wave32):
//   A (16x4 f32, 2 VGPRs):  row M = lane%16;  v.x = A[M][k0+2*(lane/16)],
//                                             v.y = A[M][k0+2*(lane/16)+1]
//   C/D (16x16 f32, 8 VGPRs): element r -> (M = r + 8*(lane/16), N = lane%16)
//   B (4x16 f32, 2 VGPRs): not tabulated in the ISA text; assumed symmetric
//     to A: col N = lane%16; v.x = B[k0+2*(lane/16)][N], v.y = next K row.
// --------------------------------------------------------------------------

// ----------------------- input projection: C = A[M,K] @ W[K,128] + b + emb
// grid.x = M/16, block = 256 (8 waves; wave w owns output cols [16w,16w+16))
__global__ void __launch_bounds__(256)
gemm_in_kernel(const float* __restrict__ A, const float* __restrict__ W,
               const float* __restrict__ bias, const float* __restrict__ emb,
               const int* __restrict__ nid, float* __restrict__ C, int K) {
  const int lane = threadIdx.x & 31;
  const int wave = threadIdx.x >> 5;
  const int half = lane >> 4;
  const int nloc = lane & 15;
  const int m0   = blockIdx.x * 16;
  const int row  = m0 + nloc;          // A-fragment row for this lane
  const int n    = wave * 16 + nloc;   // output column

  const float* Arow = A + (size_t)row * K;
  v8f acc = {};
#pragma unroll 8
  for (int k0 = 0; k0 < K; k0 += 4) {
    const int ka = k0 + 2 * half;
    v2f a; a.x = Arow[ka];
           a.y = Arow[ka + 1];
    v2f b; b.x = W[(size_t)ka * HDIM + n];
           b.y = W[(size_t)(ka + 1) * HDIM + n];
    acc = __builtin_amdgcn_wmma_f32_16x16x4_f32(false, a, false, b,
                                                (short)0, acc, false, false);
  }

  const float bn = bias[n];
#pragma unroll
  for (int r = 0; r < 8; ++r) {
    const int m = m0 + r + 8 * half;
    const float v = acc[r] + bn + emb[(size_t)nid[m] * HDIM + n];
    C[(size_t)m * HDIM + n] = v;
  }
}

// ------------- SAGE layer: Out = (S/deg) @ Wl + bl + X @ Wr  (opt. ReLU)
// The per-row mean division commutes with the K-sum, so scale after WMMA.
__global__ void __launch_bounds__(256)
sage_kernel(const float* __restrict__ S, const float* __restrict__ deg,
            const float* __restrict__ X, const float* __restrict__ Wl,
            const float* __restrict__ bl, const float* __restrict__ Wr,
            float* __restrict__ Out, int relu) {
  const int lane = threadIdx.x & 31;
  const int wave = threadIdx.x >> 5;
  const int half = lane >> 4;
  const int nloc = lane & 15;
  const int m0   = blockIdx.x * 16;
  const int row  = m0 + nloc;
  const int n    = wave * 16 + nloc;

  const float* Srow = S + (size_t)row * HDIM;
  const float* Xrow = X + (size_t)row * HDIM;
  v8f accS = {};
  v8f accX = {};
#pragma unroll 4
  for (int k0 = 0; k0 < HDIM; k0 += 4) {
    const int ka = k0 + 2 * half;
    v2f a1; a1.x = Srow[ka]; a1.y = Srow[ka + 1];
    v2f b1; b1.x = Wl[(size_t)ka * HDIM + n];
            b1.y = Wl[(size_t)(ka + 1) * HDIM + n];
    accS = __builtin_amdgcn_wmma_f32_16x16x4_f32(false, a1, false, b1,
                                                 (short)0, accS, false, false);
    v2f a2; a2.x = Xrow[ka]; a2.y = Xrow[ka + 1];
    v2f b2; b2.x = Wr[(size_t)ka * HDIM + n];
            b2.y = Wr[(size_t)(ka + 1) * HDIM + n];
    accX = __builtin_amdgcn_wmma_f32_16x16x4_f32(false, a2, false, b2,
                                                 (short)0, accX, false, false);
  }

  const float bn = bl[n];
#pragma unroll
  for (int r = 0; r < 8; ++r) {
    const int m = m0 + r + 8 * half;
    float d = deg[m];
    d = d < 1.0f ? 1.0f : d;
    float v = accS[r] / d + accX[r] + bn;
    if (relu) v = v > 0.0f ? v : 0.0f;
    Out[(size_t)m * HDIM + n] = v;
  }
}

// ----------------------- edge head: out[e] = dot(Hc[ls[e]], Hp[ld[e]])
__global__ void edge_dot_kernel(const float* __restrict__ Hc,
                                const float* __restrict__ Hp,
                                const int* __restrict__ ls,
                                const int* __restrict__ ld, int e,
                                float* __restrict__ out) {
  int wid  = (blockIdx.x * blockDim.x + threadIdx.x) >> 5;
  int lane = threadIdx.x & 31;
  if (wid >= e) return;
  int a = ls[wid];
  int b = ld[wid];
  const float4 va = *(const float4*)(Hc + (size_t)a * HDIM + lane * 4);
  const float4 vb = *(const float4*)(Hp + (size_t)b * HDIM + lane * 4);
  float v = va.x * vb.x + va.y * vb.y + va.z * vb.z + va.w * vb.w;
#pragma unroll
  for (int off = 16; off > 0; off >>= 1) v += __shfl_down(v, off, 32);
  if (lane == 0) out[wid] = v;
}

extern "C" void kernel_launch(void* const* d_in, const int* in_sizes, int n_in,
                              void* d_out, int out_size, void* d_ws,
                              size_t ws_size, hipStream_t stream) {
  (void)in_sizes; (void)n_in; (void)out_size; (void)ws_size;
  const int NC = 4096, NP = 8192, E = 500000, EL = 500000;

  const float* x_country = (const float*)d_in[0];
  const float* x_product = (const float*)d_in[1];
  const float* Wc_lin = (const float*)d_in[2];
  const float* bc_lin = (const float*)d_in[3];
  const float* Wp_lin = (const float*)d_in[4];
  const float* bp_lin = (const float*)d_in[5];
  const float* emb_c  = (const float*)d_in[6];
  const float* emb_p  = (const float*)d_in[7];
  const float* W1ep_l = (const float*)d_in[8];
  const float* b1ep   = (const float*)d_in[9];
  const float* W1ep_r = (const float*)d_in[10];
  const float* W1pc_l = (const float*)d_in[11];
  const float* b1pc   = (const float*)d_in[12];
  const float* W1pc_r = (const float*)d_in[13];
  const float* W2ep_l = (const float*)d_in[14];
  const float* b2ep   = (const float*)d_in[15];
  const float* W2ep_r = (const float*)d_in[16];
  const float* W2pc_l = (const float*)d_in[17];
  const float* b2pc   = (const float*)d_in[18];
  const float* W2pc_r = (const float*)d_in[19];
  const int* nid_c = (const int*)d_in[20];
  const int* nid_p = (const int*)d_in[21];
  const int* e_src = (const int*)d_in[22];
  const int* e_dst = (const int*)d_in[23];
  const int* l_src = (const int*)d_in[24];
  const int* l_dst = (const int*)d_in[25];

  float* ws = (float*)d_ws;
  size_t off = 0;
  float* xc   = ws + off; off += (size_t)NC * HDIM;
  float* xp   = ws + off; off += (size_t)NP * HDIM;
  float* hp   = ws + off; off += (size_t)NP * HDIM;
  float* hc   = ws + off; off += (size_t)NC * HDIM;
  float* hp2  = ws + off; off += (size_t)NP * HDIM;
  float* hc2  = ws + off; off += (size_t)NC * HDIM;
  float* sbuf = ws + off; off += (size_t)NP * HDIM;  // reused scatter buffer
  float* degp = ws + off; off += NP;
  float* degc = ws + off; off += NC;

  const int TB = 256;
  const int edge_blocks = (E * 32 + TB - 1) / TB;   // wave-per-edge kernels
  const int np_elems = NP * HDIM;
  const int nc_elems = NC * HDIM;

  // input projections (fused linear + embedding add)
  gemm_in_kernel<<<NC / 16, TB, 0, stream>>>(x_country, Wc_lin, bc_lin, emb_c,
                                             nid_c, xc, NC);
  gemm_in_kernel<<<NP / 16, TB, 0, stream>>>(x_product, Wp_lin, bp_lin, emb_p,
                                             nid_p, xp, NP);

  // degrees (edge structure is layer-invariant -> compute once)
  zero_kernel<<<64, TB, 0, stream>>>(degp, NP);
  zero_kernel<<<64, TB, 0, stream>>>(degc, NC);
  degree_kernel<<<(E + TB - 1) / TB, TB, 0, stream>>>(e_dst, E, degp);
  degree_kernel<<<(E + TB - 1) / TB, TB, 0, stream>>>(e_src, E, degc);

  // layer 1: country -> product  (hp = relu(mean(xc) @ W1ep_l + b + xp @ W1ep_r))
  zero_kernel<<<1024, TB, 0, stream>>>(sbuf, np_elems);
  scatter_kernel<<<edge_blocks, TB, 0, stream>>>(xc, e_src, e_dst, E, sbuf);
  sage_kernel<<<NP / 16, TB, 0, stream>>>(sbuf, degp, xp, W1ep_l, b1ep, W1ep_r,
                                          hp, 1);

  // layer 1: product -> country
  zero_kernel<<<1024, TB, 0, stream>>>(sbuf, nc_elems);
  scatter_kernel<<<edge_blocks, TB, 0, stream>>>(xp, e_dst, e_src, E, sbuf);
  sage_kernel<<<NC / 16, TB, 0, stream>>>(sbuf, degc, xc, W1pc_l, b1pc, W1pc_r,
                                          hc, 1);

  // layer 2: country -> product (no relu)
  zero_kernel<<<1024, TB, 0, stream>>>(sbuf, np_elems);
  scatter_kernel<<<edge_blocks, TB, 0, stream>>>(hc, e_src, e_dst, E, sbuf);
  sage_kernel<<<NP / 16, TB, 0, stream>>>(sbuf, degp, hp, W2ep_l, b2ep, W2ep_r,
                                          hp2, 0);

  // layer 2: product -> country (no relu)
  zero_kernel<<<1024, TB, 0, stream>>>(sbuf, nc_elems);
  scatter_kernel<<<edge_blocks, TB, 0, stream>>>(hp, e_dst, e_src, E, sbuf);
  sage_kernel<<<NC / 16, TB, 0, stream>>>(sbuf, degc, hc, W2pc_l, b2pc, W2pc_r,
                                          hc2, 0);

  // edge-wise dot-product classifier
  const int el_blocks = (EL * 32 + TB - 1) / TB;
  edge_dot_kernel<<<el_blocks, TB, 0, stream>>>(hc2, hp2, l_src, l_dst, EL,
                                                (float*)d_out);
}